// SkyCubeMap_56547539419766
// MI455X (gfx1250) — compile-verified
//
#include <hip/hip_runtime.h>
#include <math.h>

// Sky cubemap sampling, MI455X (gfx1250).
// Memory-bound gather kernel: cubemap (75.5 MB) is L2-resident (192 MB L2);
// acc/output are streamed with non-temporal hints. Scale-invariance of the
// cube projection lets us skip ray normalization; v_rcp_f32 replaces divides.

__device__ __forceinline__ float clamp01(float x) {
    return fminf(fmaxf(x, 0.0f), 1.0f);
}

typedef __attribute__((address_space(1))) int as1_int;
typedef __attribute__((address_space(3))) int as3_int;

__global__ __launch_bounds__(256) void sky_cube_kernel(
    const float* __restrict__ cube,   // (6, R, R, 3)
    const float* __restrict__ Km,     // (3, 3) row-major
    const float* __restrict__ Rmat,   // (3, 3) row-major
    const float* __restrict__ acc,    // (H*W)
    const int*   __restrict__ Wp,     // scalar W
    float* __restrict__ out,          // (3, H, W)
    int R, int HW)
{
    // cst[0..8] = Rm, cst[9]=fx, cst[10]=fy, cst[11]=cx, cst[12]=cy
    __shared__ float cst[16];
    const int t = threadIdx.x;

#if __has_builtin(__builtin_amdgcn_global_load_async_to_lds_b32)
    if (t < 13) {
        const float* src;
        if (t < 9) {
            src = Rmat + t;
        } else {
            const int k = (t == 9) ? 0 : (t == 10) ? 4 : (t == 11) ? 2 : 5;
            src = Km + k;
        }
        __builtin_amdgcn_global_load_async_to_lds_b32(
            (as1_int*)src, (as3_int*)&cst[t], 0, 0);
    }
#if __has_builtin(__builtin_amdgcn_s_wait_asynccnt)
    __builtin_amdgcn_s_wait_asynccnt(0);
#else
    asm volatile("s_wait_asynccnt 0" ::: "memory");
#endif
    __syncthreads();
#else
    if (t < 13) {
        float v;
        if (t < 9) {
            v = Rmat[t];
        } else {
            const int k = (t == 9) ? 0 : (t == 10) ? 4 : (t == 11) ? 2 : 5;
            v = Km[k];
        }
        cst[t] = v;
    }
    __syncthreads();
#endif

    const int idx = blockIdx.x * 256 + t;
    if (idx >= HW) return;

    const int W = *Wp;           // uniform -> scalar load
    const int x = idx % W;
    const int y = idx / W;

    // Camera-space ray (pre-rotation); normalization skipped (projection is
    // invariant under positive scaling of the direction).
    const float fx = cst[9], fy = cst[10], cx = cst[11], cy = cst[12];
    const float px = ((float)x + 0.5f - cx) * __builtin_amdgcn_rcpf(fx);
    const float py = ((float)y + 0.5f - cy) * __builtin_amdgcn_rcpf(fy);

    // world dir = dirs @ Rm : d[k] = px*Rm[0,k] + py*Rm[1,k] + Rm[2,k]
    const float dx = px * cst[0] + py * cst[3] + cst[6];
    const float dy = px * cst[1] + py * cst[4] + cst[7];
    const float dz = px * cst[2] + py * cst[5] + cst[8];

    // Face select (OpenGL convention), matching the reference's tie-breaks.
    const float ax = fabsf(dx), ay = fabsf(dy), az = fabsf(dz);
    int face; float ma, sc, tc;
    if (ax >= ay && ax >= az) {
        ma = ax; tc = -dy;
        if (dx >= 0.0f) { face = 0; sc = -dz; }
        else            { face = 1; sc =  dz; }
    } else if (ay >= az) {
        ma = ay; sc = dx;
        if (dy >= 0.0f) { face = 2; tc =  dz; }
        else            { face = 3; tc = -dz; }
    } else {
        ma = az; tc = -dy;
        if (dz >= 0.0f) { face = 4; sc =  dx; }
        else            { face = 5; sc = -dx; }
    }

    const float inv = __builtin_amdgcn_rcpf(ma);
    const float Rf  = (float)R;
    const float u   = (0.5f * (sc * inv + 1.0f)) * Rf - 0.5f;
    const float v   = (0.5f * (tc * inv + 1.0f)) * Rf - 0.5f;
    const float u0  = floorf(u), v0 = floorf(v);
    const float fu  = u - u0,    fv = v - v0;
    const float hi  = Rf - 1.0f;
    const int i0 = (int)fminf(fmaxf(u0,        0.0f), hi);
    const int i1 = (int)fminf(fmaxf(u0 + 1.0f, 0.0f), hi);
    const int j0 = (int)fminf(fmaxf(v0,        0.0f), hi);
    const int j1 = (int)fminf(fmaxf(v0 + 1.0f, 0.0f), hi);

    // 4 texel gathers; each RGB triplet is contiguous -> global_load_b96.
    const float* fb  = cube + (size_t)face * (size_t)R * (size_t)R * 3u;
    const float* r0p = fb + (size_t)j0 * (size_t)R * 3u;
    const float* r1p = fb + (size_t)j1 * (size_t)R * 3u;
    const float* p00 = r0p + 3 * i0;
    const float* p01 = r0p + 3 * i1;
    const float* p10 = r1p + 3 * i0;
    const float* p11 = r1p + 3 * i1;

    const float w00 = (1.0f - fu) * (1.0f - fv);
    const float w01 = fu * (1.0f - fv);
    const float w10 = (1.0f - fu) * fv;
    const float w11 = fu * fv;

    float r = p00[0] * w00 + p01[0] * w01 + p10[0] * w10 + p11[0] * w11;
    float g = p00[1] * w00 + p01[1] * w01 + p10[1] * w10 + p11[1] * w11;
    float b = p00[2] * w00 + p01[2] * w01 + p10[2] * w10 + p11[2] * w11;

    // Streaming data: non-temporal so it never evicts the L2-resident cubemap.
    const float a = __builtin_nontemporal_load(acc + idx);
    const bool  m = (1.0f - a) > 0.001f;
    r = m ? clamp01(r) : 0.0f;
    g = m ? clamp01(g) : 0.0f;
    b = m ? clamp01(b) : 0.0f;

    __builtin_nontemporal_store(r, out + idx);
    __builtin_nontemporal_store(g, out + HW + idx);
    __builtin_nontemporal_store(b, out + 2 * HW + idx);
}

extern "C" void kernel_launch(void* const* d_in, const int* in_sizes, int n_in,
                              void* d_out, int out_size, void* d_ws, size_t ws_size,
                              hipStream_t stream) {
    // setup_inputs order: sky_cube_map, K, Rm, T, acc, H, W
    const float* cube = (const float*)d_in[0];
    const float* Km   = (const float*)d_in[1];
    const float* Rm   = (const float*)d_in[2];
    // d_in[3] = T (unused by the eval path)
    const float* acc  = (const float*)d_in[4];
    const int*   Wp   = (const int*)d_in[6];
    float* out = (float*)d_out;

    const int HW = in_sizes[4];                                  // H*W
    const int R  = (int)lround(sqrt((double)in_sizes[0] / 18.0)); // 6*R*R*3

    const int blocks = (HW + 255) / 256;
    sky_cube_kernel<<<blocks, 256, 0, stream>>>(cube, Km, Rm, acc, Wp, out, R, HW);
}